// ChamferLoss2D_8254927143419
// MI455X (gfx1250) — compile-verified
//
#include <hip/hip_runtime.h>
#include <math.h>

// Chamfer distance 2D: N=8 batches, P=4096 points, dim=2.
// dist2[i][j] = |x_i|^2 + |y_j|^2 - 2 x_i.y_j  computed as one
// V_WMMA_F32_16X16X4_F32 per 16x16 tile (all 4 K slots used):
//   A[i,:] = [x0, x1, 1, |x|^2],  B[:,j] = [-2y0, -2y1, |y|^2, 1]^T
// Y is staged per-block in LDS; column mins merge via ds_min_u32 in LDS
// (IEEE bits of non-negative floats order as unsigned), then one global
// atomic-min pass per block merges across the 8 blocks sharing a batch.

typedef float v2f __attribute__((ext_vector_type(2)));
typedef float v8f __attribute__((ext_vector_type(8)));

#define NBATCH 8
#define NPTS   4096
#define RT_PER_WAVE 4          // 4 row tiles = 64 rows per wave
#define WAVES_PER_BATCH 64     // 64 waves * 64 rows = 4096 rows
#define TOTAL_WAVES (NBATCH * WAVES_PER_BATCH)   // 512
#define WAVES_PER_BLOCK 8
#define NBLOCKS (TOTAL_WAVES / WAVES_PER_BLOCK)  // 64 (8 blocks per batch)
#define FLT_MAX_BITS 0x7F7FFFFFu

__global__ void chamfer_init_kernel(unsigned* __restrict__ colmin) {
    int i = blockIdx.x * blockDim.x + threadIdx.x;
    if (i < NBATCH * NPTS) colmin[i] = FLT_MAX_BITS;
}

__global__ __launch_bounds__(256)
void chamfer_main_kernel(const float2* __restrict__ X,
                         const float2* __restrict__ Y,
                         unsigned* __restrict__ colmin,   // [NBATCH*NPTS] squared-dist bits
                         float* __restrict__ d1part) {    // [TOTAL_WAVES]
    __shared__ float2   sY[NPTS];      // raw Y points for this batch (32 KB)
    __shared__ unsigned sCol[NPTS];    // block-local column mins (16 KB)

    const int tid   = threadIdx.x;
    const int lane  = tid & 31;
    const int wv    = tid >> 5;                       // wave in block, 0..7
    const int wid   = blockIdx.x * WAVES_PER_BLOCK + wv;  // 0..511
    const int batch = blockIdx.x >> 3;                // 8 blocks per batch
    const int rg    = wid & 63;                       // row group within batch
    const int m0    = lane & 15;
    const bool lo   = (lane < 16);

    const float2* Xb = X + batch * NPTS;
    const float2* Yb = Y + batch * NPTS;

    // ---- stage Y into LDS, init block-local colmin ----
    for (int j = tid; j < NPTS; j += 256) {
        sY[j]   = Yb[j];
        sCol[j] = FLT_MAX_BITS;
    }
    __syncthreads();

    // ---- A fragments: 16x4 f32; lanes 0-15 hold K=0,1 ; lanes 16-31 hold K=2,3 ----
    v2f afr[RT_PER_WAVE];
    v8f racc[RT_PER_WAVE];             // running row-min (squared)
#pragma unroll
    for (int rt = 0; rt < RT_PER_WAVE; ++rt) {
        int row = rg * 64 + rt * 16 + m0;
        float2 x = Xb[row];
        afr[rt].x = lo ? x.x : 1.0f;                       // K=0 | K=2
        afr[rt].y = lo ? x.y : (x.x * x.x + x.y * x.y);    // K=1 | K=3
#pragma unroll
        for (int v = 0; v < 8; ++v) racc[rt][v] = 3.0e38f;
    }

    // ---- sweep all 256 column tiles ----
    for (int jt = 0; jt < NPTS / 16; ++jt) {
        const int col = jt * 16 + m0;
        float2 y = sY[col];
        // B fragment: 4x16 f32 — row K striped across lanes within a VGPR
        v2f bfr;
        bfr.x = lo ? (-2.0f * y.x) : (y.x * y.x + y.y * y.y);  // K=0 | K=2
        bfr.y = lo ? (-2.0f * y.y) : 1.0f;                     // K=1 | K=3

        float cmin = 3.0e38f;
#pragma unroll
        for (int rt = 0; rt < RT_PER_WAVE; ++rt) {
            v8f c = {};
            v8f d = __builtin_amdgcn_wmma_f32_16x16x4_f32(
                false, afr[rt], false, bfr, (short)0, c, false, false);
#pragma unroll
            for (int v = 0; v < 8; ++v) {
                racc[rt][v] = fminf(racc[rt][v], d[v]);  // row-min accumulate
                cmin        = fminf(cmin, d[v]);         // col-min within tile
            }
        }
        // All 32 lanes atomically min their half's partial into LDS: the two
        // lanes sharing column `col` merge in the LDS atomic unit. Clamp
        // tiny cancellation negatives so uint ordering == float ordering.
        cmin = fmaxf(cmin, 0.0f);
        atomicMin(&sCol[col], __float_as_uint(cmin));
    }

    // ---- row mins: reduce the 16 column-slot lanes within each half-wave ----
    float s = 0.0f;
#pragma unroll
    for (int rt = 0; rt < RT_PER_WAVE; ++rt) {
#pragma unroll
        for (int v = 0; v < 8; ++v) {
            float r = racc[rt][v];
            r = fminf(r, __shfl_xor(r, 1, 32));
            r = fminf(r, __shfl_xor(r, 2, 32));
            r = fminf(r, __shfl_xor(r, 4, 32));
            r = fminf(r, __shfl_xor(r, 8, 32));
            s += sqrtf(fmaxf(r, 0.0f));
        }
    }
    float tot = s + __shfl_xor(s, 16, 32);
    if (lane == 0) d1part[wid] = tot;

    // ---- merge block-local column mins into global (8 blocks per batch) ----
    __syncthreads();
    for (int j = tid; j < NPTS; j += 256)
        atomicMin(&colmin[batch * NPTS + j], sCol[j]);
}

__global__ __launch_bounds__(256)
void chamfer_final_kernel(const unsigned* __restrict__ colmin,
                          const float* __restrict__ d1part,
                          float* __restrict__ out) {
    const int b = blockIdx.x;
    const int t = threadIdx.x;
    __shared__ float sm[256];

    float s = 0.0f;
    for (int j = t; j < NPTS; j += 256)
        s += sqrtf(__uint_as_float(colmin[b * NPTS + j]));   // d2 contributions
    if (t < WAVES_PER_BATCH)
        s += d1part[b * WAVES_PER_BATCH + t];                // d1 contributions

    sm[t] = s;
    __syncthreads();
    for (int off = 128; off > 0; off >>= 1) {
        if (t < off) sm[t] += sm[t + off];
        __syncthreads();
    }
    // dist = (mean(d1) + mean(d2)) / 2 = (S1 + S2) / (2 * 4096)
    if (t == 0) out[b] = sm[0] * (1.0f / 8192.0f);
}

extern "C" void kernel_launch(void* const* d_in, const int* in_sizes, int n_in,
                              void* d_out, int out_size, void* d_ws, size_t ws_size,
                              hipStream_t stream) {
    const float2* X = (const float2*)d_in[0];
    const float2* Y = (const float2*)d_in[1];
    float* out = (float*)d_out;

    unsigned* colmin = (unsigned*)d_ws;
    float* d1part = (float*)((char*)d_ws + (size_t)NBATCH * NPTS * sizeof(unsigned));

    chamfer_init_kernel<<<(NBATCH * NPTS + 255) / 256, 256, 0, stream>>>(colmin);
    chamfer_main_kernel<<<NBLOCKS, 256, 0, stream>>>(X, Y, colmin, d1part);
    chamfer_final_kernel<<<NBATCH, 256, 0, stream>>>(colmin, d1part, out);
}